// PerScaleCalib_74131135529453
// MI455X (gfx1250) — compile-verified
//
#include <hip/hip_runtime.h>

// Problem constants (match reference)
#define BB   8
#define HH   512
#define WW   1408
#define NN   (HH * WW)        // 720896 pixels per batch image
#define BN   (BB * NN)        // 5767168 total pixels
#define EPSF 1e-6f

// Workspace layout (floats): [0,BN) num_s | [BN,2BN) den | [2BN,3BN) num_c
// Needs 3*BN*4 = ~69.2 MB of d_ws.

// ---------------------------------------------------------------------------
// Kernel 1: zero the accumulators (float4 stores, fully coalesced)
// ---------------------------------------------------------------------------
__global__ void ps_fill_zero(float4* __restrict__ p, int n4) {
    int i = blockIdx.x * blockDim.x + threadIdx.x;
    if (i < n4) p[i] = make_float4(0.f, 0.f, 0.f, 0.f);
}

// ---------------------------------------------------------------------------
// Per-pixel reprojection + bilinear scatter (only called for masked pixels)
// ---------------------------------------------------------------------------
__device__ __forceinline__ void ps_splat_pixel(
    float z, float d0, float d1, float d2, float cf,
    float fx, float fy, float cx, float cy,
    int col, int row, int bbase,
    float* __restrict__ num_s, float* __restrict__ den, float* __restrict__ num_c)
{
    // unproject (reference adds EPS to focal lengths)
    float x = z * ((float)col - cx) / (fx + EPSF);
    float y = z * ((float)row - cy) / (fy + EPSF);
    // perturb; relu on z
    float xn = x + d0;
    float yn = y + d1;
    float zn = fmaxf(z + d2, 0.f);
    if (!(zn > EPSF)) return;          // valid = (M>0) & (z_new>EPS)
    float zs = fmaxf(zn, EPSF);        // == zn here, kept for exactness
    // reproject
    float u = fx * xn / zs + cx;
    float v = fy * yn / zs + cy;
    // bilinear corner weights, computed exactly like the reference
    float u0f = floorf(u), v0f = floorf(v);
    float u1f = u0f + 1.f, v1f = v0f + 1.f;
    float us[2]  = { u0f,      u1f     };
    float vs[2]  = { v0f,      v1f     };
    float wus[2] = { u1f - u,  u - u0f };
    float wvs[2] = { v1f - v,  v - v0f };
#pragma unroll
    for (int a = 0; a < 2; ++a) {
#pragma unroll
        for (int bb2 = 0; bb2 < 2; ++bb2) {
            float uu = us[bb2], vv = vs[a];
            float wgt = wus[bb2] * wvs[a];
            if (uu >= 0.f && uu <= (float)(WW - 1) &&
                vv >= 0.f && vv <= (float)(HH - 1)) {
                int t = bbase + (int)vv * WW + (int)uu;
                // f32 atomic adds, no return -> STOREcnt path, resolved in L2
                atomicAdd(num_s + t, zn * wgt);
                atomicAdd(den   + t, wgt);
                atomicAdd(num_c + t, cf * wgt);
            }
        }
    }
}

// ---------------------------------------------------------------------------
// Kernel 2: mask scan (b128) + scatter. 4 pixels/thread, one row per quad.
// Grid: BB * 704 blocks of 256 threads  (NN = 4*256*704 exactly).
// ---------------------------------------------------------------------------
__global__ void ps_scatter(const float* __restrict__ S,
                           const float* __restrict__ M,
                           const float* __restrict__ K,
                           const float* __restrict__ D,
                           const float* __restrict__ C,
                           float* __restrict__ num_s,
                           float* __restrict__ den,
                           float* __restrict__ num_c)
{
    const int blocksPerBatch = (NN / 4) / 256;            // 704
    int b = blockIdx.x / blocksPerBatch;                   // uniform per block
    int q = (blockIdx.x % blocksPerBatch) * blockDim.x + threadIdx.x;
    int p = q * 4;                                         // pixel base in batch
    int base = b * NN + p;

    // single wide load; fully unmasked quads exit here
    float4 m4 = *reinterpret_cast<const float4*>(M + base);
    if (m4.x <= 0.f && m4.y <= 0.f && m4.z <= 0.f && m4.w <= 0.f) return;

    int dbase = b * 3 * NN + p;
    // start the 5 dependent cachelines early (global_prefetch_b8)
    __builtin_prefetch(S + base,           0, 3);
    __builtin_prefetch(C + base,           0, 3);
    __builtin_prefetch(D + dbase,          0, 3);
    __builtin_prefetch(D + dbase + NN,     0, 3);
    __builtin_prefetch(D + dbase + 2 * NN, 0, 3);

    // per-batch intrinsics: b is block-uniform -> scalar loads
    float fx = K[b * 9 + 0];
    float fy = K[b * 9 + 4];
    float cx = K[b * 9 + 2];
    float cy = K[b * 9 + 5];

    int row = p / WW;
    int col = p % WW;                                      // WW % 4 == 0: same row

    float mj[4] = { m4.x, m4.y, m4.z, m4.w };
#pragma unroll
    for (int j = 0; j < 4; ++j) {
        if (mj[j] > 0.f) {
            float z  = S[base + j];
            float d0 = D[dbase + j];
            float d1 = D[dbase + NN + j];
            float d2 = D[dbase + 2 * NN + j];
            float cf = C[base + j];
            ps_splat_pixel(z, d0, d1, d2, cf, fx, fy, cx, cy,
                           col + j, row, b * NN, num_s, den, num_c);
        }
    }
}

// ---------------------------------------------------------------------------
// Kernel 3: normalize both outputs (float4 streaming)
// d_out: [0,BN) = s_pred, [BN,2BN) = c_pred
// ---------------------------------------------------------------------------
__global__ void ps_normalize(const float4* __restrict__ ns,
                             const float4* __restrict__ dn,
                             const float4* __restrict__ nc,
                             float4* __restrict__ out)
{
    int i = blockIdx.x * blockDim.x + threadIdx.x;
    if (i >= BN / 4) return;
    float4 d = dn[i];
    float4 s = ns[i];
    float4 c = nc[i];
    float4 os, oc;
    os.x = (d.x > EPSF) ? s.x / (d.x + EPSF) : 0.f;
    os.y = (d.y > EPSF) ? s.y / (d.y + EPSF) : 0.f;
    os.z = (d.z > EPSF) ? s.z / (d.z + EPSF) : 0.f;
    os.w = (d.w > EPSF) ? s.w / (d.w + EPSF) : 0.f;
    oc.x = (d.x > EPSF) ? c.x / (d.x + EPSF) : 0.f;
    oc.y = (d.y > EPSF) ? c.y / (d.y + EPSF) : 0.f;
    oc.z = (d.z > EPSF) ? c.z / (d.z + EPSF) : 0.f;
    oc.w = (d.w > EPSF) ? c.w / (d.w + EPSF) : 0.f;
    out[i]          = os;
    out[BN / 4 + i] = oc;
}

// ---------------------------------------------------------------------------
extern "C" void kernel_launch(void* const* d_in, const int* in_sizes, int n_in,
                              void* d_out, int out_size, void* d_ws, size_t ws_size,
                              hipStream_t stream)
{
    const float* S = (const float*)d_in[0];   // S_l       [B,1,H,W]
    const float* M = (const float*)d_in[1];   // M_l       [B,1,H,W]
    const float* K = (const float*)d_in[2];   // K_l       [B,3,3]
    const float* D = (const float*)d_in[3];   // delta_xyz [B,3,H,W]
    const float* C = (const float*)d_in[4];   // conf_map  [B,1,H,W]

    float* num_s = (float*)d_ws;
    float* den   = num_s + BN;
    float* num_c = den + BN;

    // 1) zero accumulators (3*BN floats, float4 granularity)
    {
        int n4 = (3 * BN) / 4;                 // 4,325,376
        ps_fill_zero<<<(n4 + 255) / 256, 256, 0, stream>>>((float4*)d_ws, n4);
    }
    // 2) scatter splat
    {
        int blocks = BB * ((NN / 4) / 256);    // 8 * 704 = 5632
        ps_scatter<<<blocks, 256, 0, stream>>>(S, M, K, D, C, num_s, den, num_c);
    }
    // 3) normalize into d_out (s_pred then c_pred)
    {
        int n4 = BN / 4;                       // 1,441,792
        ps_normalize<<<(n4 + 255) / 256, 256, 0, stream>>>(
            (const float4*)num_s, (const float4*)den, (const float4*)num_c,
            (float4*)d_out);
    }
}